// RMS_24455543783579
// MI455X (gfx1250) — compile-verified
//
#include <hip/hip_runtime.h>
#include <math.h>

// ---------------- problem constants (from reference) ----------------
#define BATCH 2
#define HH 1024
#define WW 1024
#define HWPIX (HH * WW)
#define STEP 10
#define DH 102
#define DW 102
#define NNB 1000
#define SIGMA 256.0f            // H*0.25 == W*0.25

typedef float v2f __attribute__((ext_vector_type(2)));
typedef float v8f __attribute__((ext_vector_type(8)));

// ---------------- JAX threefry2x32 (key = [0,123]) ----------------
__device__ __forceinline__ uint32_t rotl32(uint32_t x, uint32_t n) {
    return (x << n) | (x >> (32u - n));
}
__device__ __forceinline__ void tf_round(uint32_t& x0, uint32_t& x1, uint32_t r) {
    x0 += x1; x1 = rotl32(x1, r); x1 ^= x0;
}
__device__ __forceinline__ void threefry2x32(uint32_t k0, uint32_t k1,
                                             uint32_t x0, uint32_t x1,
                                             uint32_t& o0, uint32_t& o1) {
    const uint32_t ks2 = k0 ^ k1 ^ 0x1BD11BDAu;
    x0 += k0; x1 += k1;
    tf_round(x0, x1, 13); tf_round(x0, x1, 15); tf_round(x0, x1, 26); tf_round(x0, x1, 6);
    x0 += k1; x1 += ks2 + 1u;
    tf_round(x0, x1, 17); tf_round(x0, x1, 29); tf_round(x0, x1, 16); tf_round(x0, x1, 24);
    x0 += ks2; x1 += k0 + 2u;
    tf_round(x0, x1, 13); tf_round(x0, x1, 15); tf_round(x0, x1, 26); tf_round(x0, x1, 6);
    x0 += k0; x1 += k1 + 3u;
    tf_round(x0, x1, 17); tf_round(x0, x1, 29); tf_round(x0, x1, 16); tf_round(x0, x1, 24);
    x0 += k1; x1 += ks2 + 4u;
    tf_round(x0, x1, 13); tf_round(x0, x1, 15); tf_round(x0, x1, 26); tf_round(x0, x1, 6);
    x0 += ks2; x1 += k0 + 5u;
    o0 = x0; o1 = x1;
}

// erfinv (Giles single-precision polynomial — matches XLA's f32 ErfInv path)
__device__ __forceinline__ float erfinv_f(float x) {
    float w = -log1pf(-x * x);
    float p;
    if (w < 5.0f) {
        w -= 2.5f;
        p = 2.81022636e-08f;
        p = fmaf(p, w, 3.43273939e-07f);
        p = fmaf(p, w, -3.5233877e-06f);
        p = fmaf(p, w, -4.39150654e-06f);
        p = fmaf(p, w, 0.00021858087f);
        p = fmaf(p, w, -0.00125372503f);
        p = fmaf(p, w, -0.00417768164f);
        p = fmaf(p, w, 0.246640727f);
        p = fmaf(p, w, 1.50140941f);
    } else {
        w = sqrtf(w) - 3.0f;
        p = -0.000200214257f;
        p = fmaf(p, w, 0.000100950558f);
        p = fmaf(p, w, 0.00134934322f);
        p = fmaf(p, w, -0.00367342844f);
        p = fmaf(p, w, 0.00573950773f);
        p = fmaf(p, w, -0.0076224613f);
        p = fmaf(p, w, 0.00943887047f);
        p = fmaf(p, w, 1.00167406f);
        p = fmaf(p, w, 2.83297682f);
    }
    return p * x;
}

// jax.random.normal(key(123)) element g of flattened (B,DH,DW,2,NNB) array.
__device__ __forceinline__ float jax_normal(uint32_t g, uint32_t half) {
    uint32_t x0, x1;
    const bool second = (g >= half);
    if (second) { x0 = g - half; x1 = g; } else { x0 = g; x1 = g + half; }
    uint32_t o0, o1;
    threefry2x32(0u, 123u, x0, x1, o0, o1);
    const uint32_t bits = second ? o1 : o0;
    const float f = __uint_as_float((bits >> 9) | 0x3f800000u) - 1.0f;  // [0,1)
    const float lo = -0.99999994f;                                      // nextafter(-1,0)
    const float u = fmaxf(lo, fmaf(f, 2.0f, lo));                       // hi-lo rounds to 2.0f
    return 1.41421356237f * erfinv_f(u);
}

// ---------------- sRGB -> Lab pieces ----------------
__device__ __forceinline__ float srgb_lin(float v) {
    return (v <= 0.04045f) ? v * (1.0f / 12.92f)
                           : powf((v + 0.055f) * (1.0f / 1.055f), 2.4f);
}
__device__ __forceinline__ float lab_f(float x) {
    // d = 6/29; d^3 = 0.008856452; 1/(3 d^2) = 7.787037; 4/29 = 0.13793103
    return (x > 0.008856452f) ? cbrtf(x) : fmaf(x, 7.787037f, 0.13793103f);
}

// =====================================================================
// Kernel 1: sRGB->Lab for both images. The 3x3 channel mix (with the
// white-point divide folded into the matrix rows) runs on
// V_WMMA_F32_16X16X4_F32: A = padded 16x4 matrix, B = 4x16 of linearized
// RGB for 16 pixels (built with ds_bpermute shuffles), D rows 0..2 = XYZ.
// Each wave handles 32 consecutive pixels via two WMMAs. EXEC is all-1s
// (pixel count is an exact multiple of 32; no divergent branches).
// Output: interleaved per-pixel [lab1.xyz pad | lab2.xyz pad] (32 B).
// =====================================================================
__global__ __launch_bounds__(256) void lab_wmma_kernel(
        const float* __restrict__ img1, const float* __restrict__ img2,
        float* __restrict__ labbuf) {
    const int lane = threadIdx.x & 31;
    const int waveG = (blockIdx.x * blockDim.x + threadIdx.x) >> 5;
    const int p0 = waveG * 32;              // first pixel of this wave's tile
    const int m  = lane & 15;
    const int k0 = (lane >> 4) * 2;
    const bool lolane = (lane < 16);

    // XYZ matrix with white-point divide folded into rows (row c /= WHITE[c])
    const float Mp[3][3] = {
        {0.4124564f / 0.95047f, 0.3575761f / 0.95047f, 0.1804375f / 0.95047f},
        {0.2126729f,            0.7151522f,            0.0721750f           },
        {0.0193339f / 1.08883f, 0.1191920f / 1.08883f, 0.9503041f / 1.08883f}};

    // A operand (16x4 f32): lane m=lane&15 holds K=k0,k0+1; zero padding
    v2f A;
    A.x = (m < 3 && k0 < 3)     ? Mp[m][k0]     : 0.0f;
    A.y = (m < 3 && k0 + 1 < 3) ? Mp[m][k0 + 1] : 0.0f;

    const int p  = p0 + lane;               // this lane's pixel
    const int bb = p / HWPIX;
    const int hw = p - bb * HWPIX;
    const size_t cbase = (size_t)bb * 3 * HWPIX + hw;

    float4* out4 = (float4*)labbuf;

    #pragma unroll
    for (int slot = 0; slot < 2; ++slot) {
        const float* img = (slot == 0) ? img1 : img2;
        const float lr = srgb_lin(img[cbase]);
        const float lg = srgb_lin(img[cbase + HWPIX]);
        const float lb = srgb_lin(img[cbase + 2 * HWPIX]);

        // B operands (4x16 f32): VGPR v, lanes 0-15 -> K=v, lanes 16-31 -> K=v+2
        const int i1 = m;        // sources for pixel tile 0..15
        const int i2 = 16 + m;   // sources for pixel tile 16..31
        const float r1 = __shfl(lr, i1, 32), g1 = __shfl(lg, i1, 32), b1 = __shfl(lb, i1, 32);
        const float r2 = __shfl(lr, i2, 32), g2 = __shfl(lg, i2, 32), b2 = __shfl(lb, i2, 32);
        v2f B0, B1;
        B0.x = lolane ? r1 : b1;  B0.y = lolane ? g1 : 0.0f;   // K=3 unused -> 0
        B1.x = lolane ? r2 : b2;  B1.y = lolane ? g2 : 0.0f;

        const v8f Cz = {};
        v8f D0 = __builtin_amdgcn_wmma_f32_16x16x4_f32(false, A, false, B0,
                                                       (short)0, Cz, false, false);
        v8f D1 = __builtin_amdgcn_wmma_f32_16x16x4_f32(false, A, false, B1,
                                                       (short)0, Cz, false, false);

        // D rows 0..2 (lanes 0-15, N=lane) hold X,Y,Z for that pixel tile.
        const float sx = __shfl(D1[0], m, 32);
        const float sy = __shfl(D1[1], m, 32);
        const float sz = __shfl(D1[2], m, 32);
        const float X = lolane ? D0[0] : sx;
        const float Y = lolane ? D0[1] : sy;
        const float Z = lolane ? D0[2] : sz;

        const float fx = lab_f(X), fy = lab_f(Y), fz = lab_f(Z);
        const float Lv = fmaf(116.0f, fy, -16.0f);
        const float av = 500.0f * (fx - fy);
        const float bv = 200.0f * (fy - fz);
        out4[(size_t)p * 2 + slot] = make_float4(Lv, av, bv, 0.0f);
    }
}

// =====================================================================
// Kernel 2: one block per base point. Regenerate the JAX noise, gather
// interleaved Lab pairs (L2-resident), masked RMS with deterministic
// LDS tree reduction.
// =====================================================================
__global__ __launch_bounds__(256) void rms_kernel(const float* __restrict__ labbuf,
                                                  float* __restrict__ rms) {
    const int point = blockIdx.x;                  // b*DH*DW + i*DW + j
    const int b  = point / (DH * DW);
    const int ij = point - b * (DH * DW);
    const int i  = ij / DW;
    const int j  = ij - i * DW;
    const int cy = i * STEP, cx = j * STEP;

    const float4* lab4 = (const float4*)labbuf;
    const size_t cpix = (size_t)b * HWPIX + (size_t)cy * WW + cx;
    const float4 c1 = lab4[cpix * 2 + 0];
    const float4 c2 = lab4[cpix * 2 + 1];

    const uint32_t half  = (uint32_t)((long long)BATCH * DH * DW * 2 * NNB / 2);
    const uint32_t base2 = (uint32_t)(((b * DH + i) * DW + j) * 2);

    float acc = 0.0f, cnt = 0.0f;
    for (int t = threadIdx.x; t < NNB; t += blockDim.x) {
        const uint32_t gy = base2 * NNB + (uint32_t)t;          // axis 0 (y)
        const uint32_t gx = (base2 + 1u) * NNB + (uint32_t)t;   // axis 1 (x)
        const float nny = jax_normal(gy, half);
        const float nnx = jax_normal(gx, half);
        float fy = rintf(fmaf(SIGMA, nny, (float)cy));
        float fx = rintf(fmaf(SIGMA, nnx, (float)cx));
        fy = fmaxf(fy, 0.0f);
        fx = fmaxf(fx, 0.0f);
        const int ny = (int)fy, nx = (int)fx;
        const bool valid = (ny < HH) && (nx < WW);
        const int nyc = min(ny, HH - 1), nxc = min(nx, WW - 1);

        const size_t pix = (size_t)b * HWPIX + (size_t)nyc * WW + nxc;
        const float4 n1 = lab4[pix * 2 + 0];
        const float4 n2 = lab4[pix * 2 + 1];

        const float dx1 = c1.x - n1.x, dy1 = c1.y - n1.y, dz1 = c1.z - n1.z;
        const float dx2 = c2.x - n2.x, dy2 = c2.y - n2.y, dz2 = c2.z - n2.z;
        const float con1 = sqrtf(dx1 * dx1 + dy1 * dy1 + dz1 * dz1 + 1e-12f);
        const float con2 = sqrtf(dx2 * dx2 + dy2 * dy2 + dz2 * dz2 + 1e-12f);
        const float diff = (con1 - con2) * (1.0f / 1.6f);
        if (valid) { acc += diff * diff; cnt += 1.0f; }
    }

    __shared__ float sacc[256];
    __shared__ float scnt[256];
    const int tid = threadIdx.x;
    sacc[tid] = acc; scnt[tid] = cnt;
    __syncthreads();
    for (int s = 128; s > 0; s >>= 1) {
        if (tid < s) { sacc[tid] += sacc[tid + s]; scnt[tid] += scnt[tid + s]; }
        __syncthreads();
    }
    if (tid == 0) {
        const float c = fmaxf(scnt[0], 1.0f);
        rms[point] = sqrtf(sacc[0] / c + 1e-12f);
    }
}

// =====================================================================
// Kernel 3: deterministic per-batch mean of the 10404 rms values.
// =====================================================================
__global__ __launch_bounds__(256) void mean_kernel(const float* __restrict__ rms,
                                                   float* __restrict__ out) {
    const int b = blockIdx.x;
    const int n = DH * DW;
    float s = 0.0f;
    for (int t = threadIdx.x; t < n; t += 256) s += rms[b * n + t];
    __shared__ float sm[256];
    sm[threadIdx.x] = s;
    __syncthreads();
    for (int k = 128; k > 0; k >>= 1) {
        if (threadIdx.x < k) sm[threadIdx.x] += sm[threadIdx.x + k];
        __syncthreads();
    }
    if (threadIdx.x == 0) out[b] = sm[0] / (float)n;
}

extern "C" void kernel_launch(void* const* d_in, const int* in_sizes, int n_in,
                              void* d_out, int out_size, void* d_ws, size_t ws_size,
                              hipStream_t stream) {
    const float* img1 = (const float*)d_in[0];
    const float* img2 = (const float*)d_in[1];
    float* labbuf = (float*)d_ws;                           // BATCH*HW*8 floats (67.1 MB)
    float* rmsbuf = labbuf + (size_t)BATCH * HWPIX * 8;     // BATCH*DH*DW floats

    const int totalPix = BATCH * HWPIX;                     // 2^21, multiple of 32
    const int blocks1 = totalPix / (32 * 8);                // 8 waves of 32 px per block
    lab_wmma_kernel<<<blocks1, 256, 0, stream>>>(img1, img2, labbuf);
    rms_kernel<<<BATCH * DH * DW, 256, 0, stream>>>(labbuf, rmsbuf);
    mean_kernel<<<BATCH, 256, 0, stream>>>(rmsbuf, (float*)d_out);
}